// KMeansProbSampler_11184094839231
// MI455X (gfx1250) — compile-verified
//
#include <hip/hip_runtime.h>

// ---------------------------------------------------------------------------
// KMeansProbSampler on MI455X (gfx1250, wave32)
//
// score(p,k) = |c_k|^2 - 2 p.c_k  (argmin-equivalent to dist^2), evaluated
// with V_WMMA_F32_16X16X4_F32:
//   A (16x4, per 16-pixel tile): row M = (-2r, -2c, 1, 0)
//   B (4x16, per 16-cluster grp): col N = (cr, cc, |c|^2, 0)
// B for all 8 cluster groups is prebuilt in LDS in exact per-lane register
// layout, then cached in 16 VGPRs per wave; each wave sweeps 8 pixel tiles
// (128 px) against it. Per-lane min over groups + shfl_xor reduction within
// 16-lane halves yields per-pixel argmin; the winning lane recomputes the
// exact clamped distance and scatter-adds (r*w, c*w), w = heat/dist, into an
// LDS accumulator; block flushes with global float atomics. 8 iterations are
// chained on the stream (ping-pong buffers in d_ws, last writes d_out).
// ---------------------------------------------------------------------------

typedef __attribute__((ext_vector_type(2))) float v2f;
typedef __attribute__((ext_vector_type(8))) float v8f;

#define HH 1024
#define WW 1024
#define CC 128
#define NGROUP 8            // 128 clusters / 16
#define WAVES 8             // 256-thread block = 8 wave32
#define TILES 8             // 16-pixel tiles per wave

__global__ __launch_bounds__(256) void zero_accum(float* out) {
    out[threadIdx.x] = 0.0f;   // launched with 256 threads == 2*CC
}

__global__ __launch_bounds__(256) void kmeans_step(const float* __restrict__ cs_in,
                                                   const float* __restrict__ heat,
                                                   float* __restrict__ cs_out)
{
    __shared__ float sr[CC];            // cluster row coords
    __shared__ float sc[CC];            // cluster col coords
    __shared__ v2f   sB[NGROUP * 32];   // B-matrix register image per group/lane
    __shared__ float sacc[2 * CC];      // per-block (r,c) accumulators

    const int tid  = threadIdx.x;
    const int lane = tid & 31;
    const int wave = tid >> 5;

    if (tid < CC) {
        float r = cs_in[2 * tid];
        float c = cs_in[2 * tid + 1];
        sr[tid] = r;
        sc[tid] = c;
    }
    sacc[tid] = 0.0f;                   // tid in [0,256) == 2*CC
    __syncthreads();

    // Build per-lane B register image for all 8 cluster groups.
    // B (4x16) layout: lanes 0-15 : VGPR0=B[K=0][N]=cr, VGPR1=B[K=1][N]=cc
    //                  lanes16-31 : VGPR0=B[K=2][N]=|c|^2, VGPR1=B[K=3][N]=0
    {
        const int g = tid >> 5;                 // group 0..7
        const int l = tid & 31;
        const int k = g * 16 + (l & 15);
        const float cr = sr[k];
        const float cc = sc[k];
        v2f val;
        val.x = (l < 16) ? cr : (cr * cr + cc * cc);
        val.y = (l < 16) ? cc : 0.0f;
        sB[g * 32 + l] = val;
    }
    __syncthreads();

    // Cache B for all groups in registers (8 x ds_load_b64 per wave).
    v2f vb[NGROUP];
    #pragma unroll
    for (int G = 0; G < NGROUP; ++G) vb[G] = sB[G * 32 + lane];

    const int waveBase = (blockIdx.x * WAVES + wave) * (TILES * 16);
    const int n = lane & 15;            // cluster column this lane owns in D

    for (int t = 0; t < TILES; ++t) {
        const int tileBase = waveBase + t * 16;

        // ---- A matrix: 32-bit 16x4 layout (branchless) -------------------
        // lanes 0-15 : VGPR0=A[M][K=0]=-2r, VGPR1=A[M][K=1]=-2c
        // lanes 16-31: VGPR0=A[M][K=2]=1,   VGPR1=A[M][K=3]=0
        const int p0 = tileBase + n;
        const float pr = (float)(p0 >> 10);        // row (W == 1024)
        const float pc = (float)(p0 & (WW - 1));   // col
        v2f a;
        a.x = (lane < 16) ? (-2.0f * pr) : 1.0f;
        a.y = (lane < 16) ? (-2.0f * pc) : 0.0f;

        // ---- 8 WMMAs over cluster groups, running min/argmin per lane ----
        float minv[8];
        int   mini[8];
        #pragma unroll
        for (int G = 0; G < NGROUP; ++G) {
            v8f acc = {};
            acc = __builtin_amdgcn_wmma_f32_16x16x4_f32(
                /*neg_a=*/false, a, /*neg_b=*/false, vb[G],
                /*c_mod=*/(short)0, acc, /*reuse_a=*/false, /*reuse_b=*/false);
            const int k = G * 16 + n;
            // D layout: VGPR g, lanes 0-15 -> (M=g, N=lane); 16-31 -> (M=g+8)
            #pragma unroll
            for (int g = 0; g < 8; ++g) {
                float v = acc[g];
                if (G == 0 || v < minv[g]) { minv[g] = v; mini[g] = k; }
            }
        }

        // ---- cross-lane min/argmin within 16-lane halves, then scatter ---
        #pragma unroll
        for (int g = 0; g < 8; ++g) {
            float v  = minv[g];
            int   ix = mini[g];
            #pragma unroll
            for (int m = 8; m >= 1; m >>= 1) {
                float ov = __shfl_xor(v, m, 16);
                int   oi = __shfl_xor(ix, m, 16);
                if (ov < v || (ov == v && oi < ix)) { v = ov; ix = oi; }
            }
            // lanes 0-15 hold the winner for pixel M=g, lanes 16-31 for M=g+8
            if (lane == g || lane == 16 + g) {
                const int M = (lane < 16) ? g : (g + 8);
                const int p = tileBase + M;
                const float r = (float)(p >> 10);
                const float c = (float)(p & (WW - 1));
                // exact distance (avoids cancellation in the expanded form)
                const float dr = r - sr[ix];
                const float dc = c - sc[ix];
                const float dist = fmaxf(1.0f, sqrtf(dr * dr + dc * dc));
                const float w = heat[p] / dist;
                atomicAdd(&sacc[2 * ix],     r * w);
                atomicAdd(&sacc[2 * ix + 1], c * w);
            }
        }
    }

    __syncthreads();
    atomicAdd(&cs_out[tid], sacc[tid]);   // 256 entries, one per thread
}

extern "C" void kernel_launch(void* const* d_in, const int* in_sizes, int n_in,
                              void* d_out, int out_size, void* d_ws, size_t ws_size,
                              hipStream_t stream) {
    const float* clusters = (const float*)d_in[0];   // [128,2]
    const float* heat     = (const float*)d_in[1];   // [1024,1024]
    float* out  = (float*)d_out;                     // [128,2]
    float* buf0 = (float*)d_ws;                      // ping
    float* buf1 = buf0 + 2 * CC;                     // pong

    // 16 px/tile * 8 tiles/wave * 8 waves/block = 1024 px/block
    const int nblocks = (HH * WW) / (16 * TILES * WAVES);

    const float* in = clusters;
    for (int it = 0; it < 8; ++it) {
        float* o = (it == 7) ? out : ((it & 1) ? buf1 : buf0);
        zero_accum<<<1, 256, 0, stream>>>(o);
        kmeans_step<<<nblocks, 256, 0, stream>>>(in, heat, o);
        in = o;
    }
}